// MoEExperts_32598801776958
// MI455X (gfx1250) — compile-verified
//
#include <hip/hip_runtime.h>
#include <math.h>

// Problem constants (from reference): N, H, I, E, K
#define NTOK 4096
#define HDIM 2048
#define IDIM 1408
#define NEXP 8
#define TOPK 2
#define TWO_I (2 * IDIM)
#define CAP (NTOK * TOPK)          // worst-case rows per expert
#define MTILE 32                   // rows per block (2 x 16-row WMMA frags)
#define TILES_PER_E (CAP / MTILE)  // 256 row-tiles per expert
#define HSTRIDE (IDIM + 4)         // padded LDS stride (bank-conflict free)

typedef float v2f __attribute__((ext_vector_type(2)));
typedef float v8f __attribute__((ext_vector_type(8)));

__device__ __forceinline__ float silu_fast(float g) {
    // silu(g) = g / (1 + exp(-g)) using v_exp_f32 + v_rcp_f32 (no IEEE div chain)
    return g * __builtin_amdgcn_rcpf(1.0f + __expf(-g));
}

// ---------------------------------------------------------------------------
// Kernel 1: zero output accumulator and per-expert dispatch counters
// ---------------------------------------------------------------------------
__global__ void MoE_zero_kernel(float* __restrict__ out, int* __restrict__ counts) {
    size_t i = (size_t)blockIdx.x * blockDim.x + threadIdx.x;
    size_t total = (size_t)NTOK * HDIM;
    size_t stride = (size_t)gridDim.x * blockDim.x;
    for (size_t j = i; j < total; j += stride) out[j] = 0.0f;
    if (i < NEXP) counts[i] = 0;
}

// ---------------------------------------------------------------------------
// Kernel 2: dispatch — build per-expert (token, weight) lists
// ---------------------------------------------------------------------------
__global__ void MoE_dispatch_kernel(const int* __restrict__ sel,
                                    const float* __restrict__ rw,
                                    int* __restrict__ counts,
                                    int* __restrict__ toks,
                                    float* __restrict__ wgts) {
    int p = blockIdx.x * blockDim.x + threadIdx.x;
    if (p >= NTOK * TOPK) return;
    int e = sel[p];
    float w = rw[p];
    int slot = atomicAdd(&counts[e], 1);
    toks[e * CAP + slot] = p / TOPK;   // token id
    wgts[e * CAP + slot] = w;
}

// ---------------------------------------------------------------------------
// Kernel 3: fused per-expert SwiGLU FFN on 32-token tiles using f32 WMMA.
//   Two 16-row A-fragments share every B-operand load (halves L2 B-traffic).
//   gu = X_tile @ Wgu ; h = silu(gate)*up (kept in LDS) ; y = h @ Wd ;
//   out[token] += routing_w * y   (global_atomic_add_f32)
// ---------------------------------------------------------------------------
__global__ __launch_bounds__(256)
void MoE_fused_kernel(const float* __restrict__ X,
                      const float* __restrict__ Wgu_all,
                      const float* __restrict__ Wd_all,
                      const int*   __restrict__ counts,
                      const int*   __restrict__ toks,
                      const float* __restrict__ wgts,
                      float*       __restrict__ out) {
    __shared__ float h_s[MTILE * HSTRIDE];   // 32 x 1408 intermediate (+pad)
    __shared__ int   tok_s[MTILE];
    __shared__ float w_s[MTILE];

    const int e    = blockIdx.x / TILES_PER_E;
    const int tile = blockIdx.x % TILES_PER_E;
    const int cnt  = counts[e];
    const int base = tile * MTILE;
    if (base >= cnt) return;              // uniform exit: EXEC stays all-1s

    if (threadIdx.x < MTILE) {
        int slot = base + threadIdx.x;
        if (slot < cnt) {
            tok_s[threadIdx.x] = toks[e * CAP + slot];
            w_s[threadIdx.x]   = wgts[e * CAP + slot];
        } else {
            tok_s[threadIdx.x] = 0;       // valid row, weight 0 -> no effect
            w_s[threadIdx.x]   = 0.0f;
        }
    }
    __syncthreads();

    const int lane = threadIdx.x & 31;
    const int wv   = threadIdx.x >> 5;    // wave id 0..7
    const int hh   = lane >> 4;           // lane half: 0 or 1
    const int mcol = lane & 15;           // row index for A, col index for B/C/D
    const int koff = hh << 1;             // K offset: lanes16-31 hold K=2,3

    const float* __restrict__ Wgu = Wgu_all + (size_t)e * HDIM * TWO_I;
    const float* __restrict__ Wd  = Wd_all  + (size_t)e * IDIM * HDIM;
    // A-operands (16x4 f32 layout): lane holds row mcol, K = koff, koff+1
    const float* __restrict__ xrow0 = X + (size_t)tok_s[mcol]      * HDIM + koff;
    const float* __restrict__ xrow1 = X + (size_t)tok_s[16 + mcol] * HDIM + koff;

    // ---------------- Phase A: gate_up GEMM + SwiGLU -> h_s ----------------
    for (int ci = 0; ci < IDIM / 16 / 8; ++ci) {          // 11 col-tiles/wave
        const int i0 = (wv + 8 * ci) * 16;
        v8f accG0 = {}, accG1 = {};
        v8f accU0 = {}, accU1 = {};
        // B-operand (4x16 f32): lane holds col mcol, rows koff, koff+1
        const float* bg = Wgu + (size_t)koff * TWO_I + i0 + mcol;
        const float* bu = bg + IDIM;
        #pragma unroll 4
        for (int k0 = 0; k0 < HDIM; k0 += 4) {
            v2f A0 = *(const v2f*)(xrow0 + k0);           // b64 loads
            v2f A1 = *(const v2f*)(xrow1 + k0);
            const float* bgk = bg + (size_t)k0 * TWO_I;
            const float* buk = bu + (size_t)k0 * TWO_I;
            v2f Bg; Bg[0] = bgk[0]; Bg[1] = bgk[TWO_I];
            v2f Bu; Bu[0] = buk[0]; Bu[1] = buk[TWO_I];
            accG0 = __builtin_amdgcn_wmma_f32_16x16x4_f32(
                false, A0, false, Bg, (short)0, accG0, false, false);
            accG1 = __builtin_amdgcn_wmma_f32_16x16x4_f32(
                false, A1, false, Bg, (short)0, accG1, false, false);
            accU0 = __builtin_amdgcn_wmma_f32_16x16x4_f32(
                false, A0, false, Bu, (short)0, accU0, false, false);
            accU1 = __builtin_amdgcn_wmma_f32_16x16x4_f32(
                false, A1, false, Bu, (short)0, accU1, false, false);
        }
        // C/D layout: VGPR v holds rows v (lanes 0-15) and v+8 (lanes 16-31)
        #pragma unroll
        for (int v = 0; v < 8; ++v) {
            int r = v + hh * 8;
            h_s[r * HSTRIDE + i0 + mcol]              = silu_fast(accG0[v]) * accU0[v];
            h_s[(16 + r) * HSTRIDE + i0 + mcol]       = silu_fast(accG1[v]) * accU1[v];
        }
    }
    __syncthreads();

    // ---------------- Phase B: down GEMM + weighted combine ----------------
    const float* hrow0 = &h_s[mcol * HSTRIDE + koff];         // A from LDS
    const float* hrow1 = &h_s[(16 + mcol) * HSTRIDE + koff];
    for (int ci = 0; ci < HDIM / 16 / 8; ++ci) {          // 16 col-tiles/wave
        const int c0 = (wv + 8 * ci) * 16;
        v8f acc0 = {}, acc1 = {};
        const float* bd = Wd + (size_t)koff * HDIM + c0 + mcol;
        #pragma unroll 4
        for (int k0 = 0; k0 < IDIM; k0 += 4) {
            v2f A0 = *(const v2f*)(hrow0 + k0);           // ds_load_b64
            v2f A1 = *(const v2f*)(hrow1 + k0);
            const float* bdk = bd + (size_t)k0 * HDIM;
            v2f Bd; Bd[0] = bdk[0]; Bd[1] = bdk[HDIM];
            acc0 = __builtin_amdgcn_wmma_f32_16x16x4_f32(
                false, A0, false, Bd, (short)0, acc0, false, false);
            acc1 = __builtin_amdgcn_wmma_f32_16x16x4_f32(
                false, A1, false, Bd, (short)0, acc1, false, false);
        }
        #pragma unroll
        for (int v = 0; v < 8; ++v) {
            int r = v + hh * 8;
            float w0 = w_s[r];
            if (w0 != 0.0f) {
                unsafeAtomicAdd(&out[(size_t)tok_s[r] * HDIM + c0 + mcol],
                                acc0[v] * w0);
            }
            float w1 = w_s[16 + r];
            if (w1 != 0.0f) {
                unsafeAtomicAdd(&out[(size_t)tok_s[16 + r] * HDIM + c0 + mcol],
                                acc1[v] * w1);
            }
        }
    }
}

// ---------------------------------------------------------------------------
extern "C" void kernel_launch(void* const* d_in, const int* in_sizes, int n_in,
                              void* d_out, int out_size, void* d_ws, size_t ws_size,
                              hipStream_t stream) {
    const float* X   = (const float*)d_in[0];   // hidden_states (N,H) f32
    const float* rw  = (const float*)d_in[1];   // routing_weights (N,K) f32
    const int*   sel = (const int*)  d_in[2];   // selected_experts (N,K) i32
    const float* Wgu = (const float*)d_in[3];   // gate_up_proj (E,H,2I) f32
    const float* Wd  = (const float*)d_in[4];   // down_proj (E,I,H) f32
    float* out = (float*)d_out;

    // workspace layout: counts | token lists | weight lists  (~0.5 MB)
    int*   counts = (int*)d_ws;
    int*   toks   = (int*)((char*)d_ws + 1024);
    float* wgts   = (float*)((char*)d_ws + 1024 + sizeof(int) * NEXP * CAP);

    MoE_zero_kernel<<<2048, 256, 0, stream>>>(out, counts);
    MoE_dispatch_kernel<<<(NTOK * TOPK + 255) / 256, 256, 0, stream>>>(
        sel, rw, counts, toks, wgts);
    MoE_fused_kernel<<<NEXP * TILES_PER_E, 256, 0, stream>>>(
        X, Wgu, Wd, counts, toks, wgts, out);
}